// MF_35175782154367
// MI455X (gfx1250) — compile-verified
//
#include <hip/hip_runtime.h>
#include <hip/hip_bf16.h>
#include <stdint.h>

#ifndef __has_builtin
#define __has_builtin(x) 0
#endif

// ---------------- geometry ----------------
#define IMG_H   256
#define IMG_W   256
#define PLANES  48          // 16 batches * 3 channels
#define KRAD    3           // 7x7 median
constexpr int TW       = 32;            // output tile width
constexpr int TH       = 8;             // output tile height
constexpr int HALO_H   = TH + 2 * KRAD; // 14 rows
constexpr int LSTRIDE  = 40;            // floats per LDS row = 10 x float4 chunks
constexpr int CHUNKS_X = LSTRIDE / 4;   // 10 16-byte chunks per halo row
// tile column 0 corresponds to global x = tx0 - 4 (16B-aligned since tx0 % 32 == 0)

// ---------------- LDS helpers ----------------
typedef __attribute__((address_space(3))) float lds_float_t;

__device__ __forceinline__ uint32_t lds_byte_addr(float* p) {
    // generic -> LDS address space -> 32-bit LDS byte address
    return (uint32_t)(uintptr_t)(lds_float_t*)p;
}

// Async 16-byte copy from global memory into LDS (per-lane, EXEC-masked).
// Tracked by ASYNCcnt; must be followed by s_wait_asynccnt before LDS reads.
__device__ __forceinline__ void async_g2l_b128(const float* gsrc, float* ldst) {
    uint32_t laddr = lds_byte_addr(ldst);
    asm volatile("global_load_async_to_lds_b128 %0, %1, off"
                 :: "v"(laddr), "v"(gsrc)
                 : "memory");
}

__device__ __forceinline__ void wait_async_zero() {
#if __has_builtin(__builtin_amdgcn_s_wait_asynccnt)
    __builtin_amdgcn_s_wait_asynccnt(0);
#else
    asm volatile("s_wait_asynccnt 0" ::: "memory");
#endif
}

// ---------------- compare-exchange networks ----------------
__device__ __forceinline__ void cas(float& a, float& b) {
    float mn = fminf(a, b);
    float mx = fmaxf(a, b);
    a = mn; b = mx;
}

// Binary-tree min-reduction over indices {BASE, BASE+2, ..., TOP} -> min at v[BASE].
// Touches only those indices; multiset preserved; depth log2(count).
template<int BASE, int TOP>
__device__ __forceinline__ void tree_min(float* v) {
    #pragma unroll
    for (int step = 2; BASE + step <= TOP; step <<= 1) {
        #pragma unroll
        for (int i = BASE; i + step <= TOP; i += 2 * step) cas(v[i], v[i + step]);
    }
}

// Binary-tree max-reduction over indices {TOP, TOP-2, ..., BOT} -> max at v[TOP].
template<int TOP, int BOT>
__device__ __forceinline__ void tree_max(float* v) {
    #pragma unroll
    for (int step = 2; TOP - step >= BOT; step <<= 1) {
        #pragma unroll
        for (int i = TOP; i - step >= BOT; i -= 2 * step) cas(v[i - step], v[i]);
    }
}

// After this: v[0] = min of v[0..N-1], v[N-1] = max of v[0..N-1].
// Multiset preserved. Op count = ceil(3N/2)-2 (optimal); depth ~ log2(N)+1.
template<int N>
__device__ __forceinline__ void minmaxN(float* v) {
    // pair phase: mins land on even slots, maxes on odd slots (v[N-1] unpaired if N odd)
    #pragma unroll
    for (int i = 0; i + 1 < N; i += 2) cas(v[i], v[i + 1]);
    if constexpr ((N & 1) == 0) {
        tree_min<0, N - 2>(v);      // min over even slots -> v[0]
        tree_max<N - 1, 1>(v);      // max over odd slots  -> v[N-1]
    } else {
        tree_min<0, N - 3>(v);      // min over paired even slots -> v[0]
        cas(v[0], v[N - 1]);        // fold unpaired element into min; loser stays high
        tree_max<N - 2, 1>(v);      // max over odd slots -> v[N-2]
        cas(v[N - 2], v[N - 1]);    // overall max -> v[N-1]
    }
}

// v[0..N-1] holds an odd-size candidate set whose middle element is the answer.
// Repeatedly eject min & max and recurse on the interior.
template<int N>
__device__ __forceinline__ float shrink_median(float* v) {
    minmaxN<N>(v);
    if constexpr (N <= 3) {
        return v[1];
    } else {
        return shrink_median<N - 2>(v + 1);
    }
}

// ---------------- kernel ----------------
__global__ void __launch_bounds__(TW * TH)
MF_35175782154367_median7(const float* __restrict__ img, float* __restrict__ out) {
    __shared__ alignas(16) float tile[HALO_H * LSTRIDE];

    const int tid   = threadIdx.x;
    const int lx    = tid & (TW - 1);
    const int ly    = tid / TW;
    const int tx0   = blockIdx.x * TW;
    const int ty0   = blockIdx.y * TH;
    const int plane = blockIdx.z;

    const float* src = img + (size_t)plane * (IMG_W * IMG_H);

    // ---- stage halo tile (14 rows x 40 floats) into LDS as 16B async chunks ----
    // Image width % 4 == 0 and chunks are 4-aligned in x, so every chunk is either
    // fully in-bounds (async b128 copy) or fully padding (-1.0 in raw domain,
    // which is the image of kornia's zero pad in [0,1] space).
    if (tid < HALO_H * CHUNKS_X) {              // 140 chunks, < 1 per thread
        const int hy  = tid / CHUNKS_X;
        const int cx  = tid - hy * CHUNKS_X;
        const int gy  = ty0 + hy - KRAD;
        const int gx0 = tx0 + cx * 4 - 4;       // 16B-aligned global column
        float* dst = &tile[hy * LSTRIDE + cx * 4];
        const bool inb = (gy >= 0) & (gy < IMG_H) & (gx0 >= 0) & (gx0 + 3 < IMG_W);
        if (inb) {
            async_g2l_b128(src + (size_t)gy * IMG_W + gx0, dst);
        } else {
            *(float4*)dst = make_float4(-1.0f, -1.0f, -1.0f, -1.0f);
        }
    }
    wait_async_zero();     // my wave's async LDS writes are done
    __syncthreads();       // everyone's writes visible

    // ---- forgetful selection: median (rank 24) of the 49 window values ----
    // window cols gx in [tx0+lx-3, tx0+lx+3]  ->  tile cols [lx+1, lx+7]
    const float* wp = &tile[ly * LSTRIDE + lx + 1];
    auto wval = [&](int k) -> float {            // k = 0..48, const-folded when unrolled
        return wp[(k / 7) * LSTRIDE + (k % 7)];
    };

    float v[26];
    #pragma unroll
    for (int k = 0; k < 26; ++k) v[k] = wval(k);

    int next = 26;
    #pragma unroll
    for (int it = 0; it < 11; ++it) {
        minmaxN<26>(v);        // v[0]=min (can't be median), v[25]=max (can't be median)
        v[0]  = wval(next++);  // overwrite ejected min
        v[25] = wval(next++);  // overwrite ejected max
    }
    minmaxN<26>(v);
    v[0] = wval(48);           // last element replaces ejected min; stale max at v[25] excluded

    // 25 live candidates in v[0..24]; global median is their middle element.
    const float med = shrink_median<25>(v);

    // the pre/post affines cancel exactly; only the final clip survives
    const float r = fminf(1.0f, fmaxf(-1.0f, med));

    const int gx = tx0 + lx;
    const int gy = ty0 + ly;
    out[(size_t)plane * (IMG_W * IMG_H) + gy * IMG_W + gx] = r;
}

// ---------------- launch ----------------
extern "C" void kernel_launch(void* const* d_in, const int* in_sizes, int n_in,
                              void* d_out, int out_size, void* d_ws, size_t ws_size,
                              hipStream_t stream) {
    (void)in_sizes; (void)n_in; (void)d_ws; (void)ws_size; (void)out_size;
    const float* image = (const float*)d_in[0];   // d_in[1] (cover_image) is unused
    float* out = (float*)d_out;

    dim3 grid(IMG_W / TW, IMG_H / TH, PLANES);    // (8, 32, 48)
    dim3 block(TW * TH);                          // 256 threads = 8 wave32
    MF_35175782154367_median7<<<grid, block, 0, stream>>>(image, out);
}